// LSTMTracker_29480655519876
// MI455X (gfx1250) — compile-verified
//
#include <hip/hip_runtime.h>

// ---------------------------------------------------------------------------
// Fused LSTM tracker for gfx1250 (MI455X), wave32 + V_WMMA_F32_16X16X4_F32.
//
//   prep kernel : Wc = W_ih @ W_emb  (80x12),  b_tot = W_ih@b_emb + b_ih + b_hh
//   main kernel : per 16-row batch tile, loop t=0..511:
//                   gates = bias + X_t(16x12)@Wc.T + h(16x20)@W_hh.T   (WMMA)
//                   LSTM cell elementwise (v_tanh_f32), output head 16x3.
// ---------------------------------------------------------------------------

typedef __attribute__((ext_vector_type(2))) float v2f;
typedef __attribute__((ext_vector_type(4))) float v4f;
typedef __attribute__((ext_vector_type(8))) float v8f;

#define T_STEPS 512
#define BATCH   2048
#define IN_DIM  12
#define E_DIM   15
#define H_DIM   20
#define NG      80      // 4*H
#define GSTRIDE 84      // padded LDS row stride for the 16x80 gate tile

__device__ __forceinline__ v8f wmma_f32_16x16x4(v2f a, v2f b, v8f c) {
  // 8-arg form: (neg_a, A, neg_b, B, c_mod, C, reuse_a, reuse_b)
  return __builtin_amdgcn_wmma_f32_16x16x4_f32(false, a, false, b,
                                               (short)0, c, false, false);
}

// Fast transcendentals on the recurrence critical path: gfx1250 has
// V_TANH_F32 (TRANS32). sigmoid(x) = 0.5 + 0.5*tanh(0.5x) -> 1 trans + 1 fma.
__device__ __forceinline__ float tanh_f(float x) {
#if __has_builtin(__builtin_amdgcn_tanhf)
  return __builtin_amdgcn_tanhf(x);
#else
  // fallback: 2/(1+exp(-2x)) - 1 with raw v_rcp_f32 (no IEEE divide chain)
  return 2.0f * __builtin_amdgcn_rcpf(1.0f + __expf(-2.0f * x)) - 1.0f;
#endif
}
__device__ __forceinline__ float sigmoid_f(float x) {
  return fmaf(0.5f, tanh_f(0.5f * x), 0.5f);
}

// ---------------------------------------------------------------------------
// Prep: fold the embedding into the input projection (tiny, one-shot).
// ---------------------------------------------------------------------------
__global__ void lstm_prep(const float* __restrict__ W_emb,
                          const float* __restrict__ b_emb,
                          const float* __restrict__ W_ih,
                          const float* __restrict__ b_ih,
                          const float* __restrict__ b_hh,
                          float* __restrict__ Wc,      // 80x12
                          float* __restrict__ b_tot) { // 80
  int i = blockIdx.x * blockDim.x + threadIdx.x;
  if (i < NG * IN_DIM) {
    int n = i / IN_DIM, k = i % IN_DIM;
    float s = 0.0f;
    for (int e = 0; e < E_DIM; ++e)
      s += W_ih[n * E_DIM + e] * W_emb[e * IN_DIM + k];
    Wc[i] = s;
  } else if (i < NG * IN_DIM + NG) {
    int n = i - NG * IN_DIM;
    float s = b_ih[n] + b_hh[n];
    for (int e = 0; e < E_DIM; ++e)
      s += W_ih[n * E_DIM + e] * b_emb[e];
    b_tot[n] = s;
  }
}

// ---------------------------------------------------------------------------
// Fused recurrence. Block = 5 waves (160 threads) = one 16-row batch tile.
// Wave w computes gate columns [16w, 16w+16) with 8 WMMAs per timestep.
// ---------------------------------------------------------------------------
__global__ __launch_bounds__(160) void lstm_fused(
    const float* __restrict__ X,      // (T, B, 12)
    const float* __restrict__ W_hh,   // (80, 20)
    const float* __restrict__ W_out,  // (3, 20)
    const float* __restrict__ b_out,  // (3)
    const float* __restrict__ Wc,     // (80, 12) from prep
    const float* __restrict__ b_tot,  // (80)     from prep
    float* __restrict__ out) {        // (T, B, 3)
  __shared__ alignas(16) float s_gates[16 * GSTRIDE];
  __shared__ alignas(16) float s_h[16 * H_DIM];
  __shared__ alignas(16) float s_wout[3 * H_DIM];
  __shared__ float s_bout[3];

  const int tid  = threadIdx.x;
  const int lane = tid & 31;
  const int wv   = tid >> 5;        // 0..4 : N-tile owned by this wave
  const int half = lane >> 4;       // 0/1  : selects K pair in A/B fragments
  const int l16  = lane & 15;
  const int row0 = blockIdx.x * 16; // batch tile base

  // one-time: output head weights to LDS, zero h state
  if (tid < 60) s_wout[tid] = W_out[tid];
  if (tid >= 60 && tid < 63) s_bout[tid - 60] = b_out[tid - 60];
  s_h[tid]       = 0.0f;            // 320 floats / 160 threads
  s_h[tid + 160] = 0.0f;

  // gate column handled by this lane, and its (folded) bias
  const int n = wv * 16 + l16;
  const float bias = b_tot[n];

  // B fragments (resident in VGPRs for all 512 steps).
  // Layout: VGPR pair {x,y} = K{0,1} for lanes 0-15, K{2,3} for lanes 16-31.
  v2f whh[5];
#pragma unroll
  for (int ks = 0; ks < 5; ++ks) {
    int k = ks * 4 + half * 2;
    whh[ks].x = W_hh[n * H_DIM + k];
    whh[ks].y = W_hh[n * H_DIM + k + 1];
  }
  v2f wcf[3];
#pragma unroll
  for (int ks = 0; ks < 3; ++ks) {
    int k = ks * 4 + half * 2;
    wcf[ks].x = Wc[n * IN_DIM + k];
    wcf[ks].y = Wc[n * IN_DIM + k + 1];
  }

  // elementwise cell mapping: 2 (m,h) cells per thread (320 = 2*160)
  const int e0 = tid * 2;
  const int em = e0 / H_DIM;        // row in tile
  const int eh = e0 % H_DIM;        // hidden idx (even)
  float c0 = 0.0f, c1 = 0.0f;

  // A-fragment row for this lane
  const int am = l16;

  // X fragments for t=0 (double-buffered across the loop)
  v2f xf[3];
  {
    const float* xp = X + ((size_t)row0 + am) * IN_DIM;
#pragma unroll
    for (int f = 0; f < 3; ++f) {
      int k = f * 4 + half * 2;
      xf[f].x = xp[k];
      xf[f].y = xp[k + 1];
    }
  }
  __syncthreads();

  for (int t = 0; t < T_STEPS; ++t) {
    // h A-fragments from LDS (8B-aligned pair loads)
    v2f hf[5];
#pragma unroll
    for (int f = 0; f < 5; ++f) {
      int k = f * 4 + half * 2;
      v2f v;
      v.x = s_h[am * H_DIM + k];
      v.y = s_h[am * H_DIM + k + 1];
      hf[f] = v;
    }

    v8f acc;
#pragma unroll
    for (int r = 0; r < 8; ++r) acc[r] = bias;

#pragma unroll
    for (int f = 0; f < 3; ++f) acc = wmma_f32_16x16x4(xf[f], wcf[f], acc);
#pragma unroll
    for (int f = 0; f < 5; ++f) acc = wmma_f32_16x16x4(hf[f], whh[f], acc);

    // prefetch next timestep's X fragments while gates settle
    if (t + 1 < T_STEPS) {
      const float* xp = X + ((size_t)(t + 1) * BATCH + row0 + am) * IN_DIM;
#pragma unroll
      for (int f = 0; f < 3; ++f) {
        int k = f * 4 + half * 2;
        xf[f].x = xp[k];
        xf[f].y = xp[k + 1];
      }
    }

    // spill gate tile to LDS in WMMA C layout: row = r + 8*half, col = n
#pragma unroll
    for (int r = 0; r < 8; ++r) {
      int m = r + half * 8;
      s_gates[m * GSTRIDE + n] = acc[r];
    }
    __syncthreads();

    // LSTM cell update: 2 cells per thread (gate order i,f,g,o)
    {
      const float* gm = s_gates + em * GSTRIDE;
      float i0 = sigmoid_f(gm[eh]);
      float f0 = sigmoid_f(gm[20 + eh]);
      float g0 = tanh_f(gm[40 + eh]);
      float o0 = sigmoid_f(gm[60 + eh]);
      c0 = f0 * c0 + i0 * g0;
      float h0 = o0 * tanh_f(c0);

      float i1 = sigmoid_f(gm[eh + 1]);
      float f1 = sigmoid_f(gm[21 + eh]);
      float g1 = tanh_f(gm[41 + eh]);
      float o1 = sigmoid_f(gm[61 + eh]);
      c1 = f1 * c1 + i1 * g1;
      float h1 = o1 * tanh_f(c1);

      s_h[em * H_DIM + eh] = h0;
      s_h[em * H_DIM + eh + 1] = h1;
    }
    __syncthreads();

    // output head: 48 threads, each one (row, out-channel) dot of length 20,
    // vectorized as 5x float4 LDS reads per operand (rows are 80B-aligned).
    if (tid < 48) {
      int m = tid / 3, j = tid % 3;
      const v4f* hr = (const v4f*)(s_h + m * H_DIM);
      const v4f* wr = (const v4f*)(s_wout + j * H_DIM);
      float s = s_bout[j];
#pragma unroll
      for (int k = 0; k < 5; ++k) {
        v4f a = hr[k], b = wr[k];
        s = fmaf(a.x, b.x, s);
        s = fmaf(a.y, b.y, s);
        s = fmaf(a.z, b.z, s);
        s = fmaf(a.w, b.w, s);
      }
      out[((size_t)t * BATCH + row0 + m) * 3 + j] = s;
    }
  }
}

// ---------------------------------------------------------------------------
extern "C" void kernel_launch(void* const* d_in, const int* in_sizes, int n_in,
                              void* d_out, int out_size, void* d_ws,
                              size_t ws_size, hipStream_t stream) {
  const float* X     = (const float*)d_in[0];  // (512, 2048, 12)
  const float* W_emb = (const float*)d_in[1];  // (15, 12)
  const float* b_emb = (const float*)d_in[2];  // (15)
  const float* W_ih  = (const float*)d_in[3];  // (80, 15)
  const float* b_ih  = (const float*)d_in[4];  // (80)
  const float* W_hh  = (const float*)d_in[5];  // (80, 20)
  const float* b_hh  = (const float*)d_in[6];  // (80)
  const float* W_out = (const float*)d_in[7];  // (3, 20)
  const float* b_out = (const float*)d_in[8];  // (3)
  float* out = (float*)d_out;

  float* Wc    = (float*)d_ws;          // 960 floats
  float* b_tot = Wc + NG * IN_DIM;      // 80 floats

  lstm_prep<<<5, 256, 0, stream>>>(W_emb, b_emb, W_ih, b_ih, b_hh, Wc, b_tot);
  lstm_fused<<<BATCH / 16, 160, 0, stream>>>(X, W_hh, W_out, b_out, Wc, b_tot,
                                             out);
}